// DyGraphConv2d_6296422056173
// MI455X (gfx1250) — compile-verified
//
#include <hip/hip_runtime.h>

typedef __attribute__((ext_vector_type(2))) float v2f;
typedef __attribute__((ext_vector_type(8))) float v8f;

#define WMMA_F32(a, b, c) \
  __builtin_amdgcn_wmma_f32_16x16x4_f32(false, (a), false, (b), (short)0, (c), false, false)

// Wave-local LDS fence: all of this wave's DS ops complete (DScnt==0) before
// proceeding. LDS ops from one wave are processed in-order, so this makes the
// wave's own cross-lane LDS writes visible to its subsequent reads without a
// block-wide barrier. "memory" clobber = compiler-level fence.
#define WAVE_LDS_FENCE() asm volatile("s_wait_dscnt 0x0" ::: "memory")

constexpr int B_ = 8;      // batch
constexpr int C_ = 64;     // channels
constexpr int N_ = 4096;   // points
constexpr int O_ = 128;    // output channels
constexpr int K_ = 9;      // neighbors

// ---------------------------------------------------------------------------
// Kernel 1: x2[b,n] = sum_c x[b,c,n]^2
// ---------------------------------------------------------------------------
__global__ __launch_bounds__(256) void k_x2(const float* __restrict__ x,
                                            float* __restrict__ x2) {
  int t = blockIdx.x * 256 + threadIdx.x;        // [0, B*N)
  int b = t >> 12;
  int n = t & (N_ - 1);
  const float* xp = x + (size_t)b * C_ * N_ + n;
  float s = 0.f;
#pragma unroll
  for (int c = 0; c < C_; ++c) {
    float v = xp[(size_t)c * N_];
    s += v * v;
  }
  x2[t] = s;
}

// ---------------------------------------------------------------------------
// Kernel 2: fused Gram-matrix (fp32 WMMA) + streaming top-9 per row.
// Score s(i,j) = 2*inner(i,j) - x2[j]  (same per-row ordering as neg_dist).
// Block = 8 independent waves; each wave owns 32 query rows (two 16x16
// accumulators) and sweeps all 256 column tiles. Scores round-trip through a
// per-wave padded LDS tile for the per-row register top-k scan; wave-local
// s_wait_dscnt replaces block barriers (LDS region is private per wave).
// ---------------------------------------------------------------------------
__global__ __launch_bounds__(256) void k_knn(const float* __restrict__ x,
                                             const float* __restrict__ x2,
                                             int* __restrict__ nn) {
  __shared__ float sm[8][32 * 17];               // 17-float pitch: conflict-free
  const int lane = threadIdx.x & 31;
  const int wave = threadIdx.x >> 5;
  const int b    = blockIdx.y;
  const int n0   = blockIdx.x * 256 + wave * 32; // 32 query rows for this wave
  const int lm   = lane & 15;
  const int coff = (lane < 16) ? 0 : 2;          // A/B fp32 16x4 lane->K mapping
  const float* xb = x + (size_t)b * C_ * N_;

  // Preload A tiles (query rows), kept in VGPRs across the whole column sweep.
  float a1[32], a2[32];
#pragma unroll
  for (int kk = 0; kk < 16; ++kk) {
    int c = 4 * kk + coff;
    a1[2 * kk + 0] = xb[(size_t)(c + 0) * N_ + n0 + lm];
    a1[2 * kk + 1] = xb[(size_t)(c + 1) * N_ + n0 + lm];
    a2[2 * kk + 0] = xb[(size_t)(c + 0) * N_ + n0 + 16 + lm];
    a2[2 * kk + 1] = xb[(size_t)(c + 1) * N_ + n0 + 16 + lm];
  }

  float tv[K_];
  int   ti[K_];
#pragma unroll
  for (int q = 0; q < K_; ++q) { tv[q] = -1e30f; ti[q] = 0; }

  float* smw = sm[wave];
  const int rbase = (lane < 16) ? 0 : 8;         // C/D fp32 16x16 lane->M mapping

  for (int m0 = 0; m0 < N_; m0 += 16) {
    v8f acc1 = {};
    v8f acc2 = {};
    const int mcol = m0 + lm;
#pragma unroll
    for (int kk = 0; kk < 16; ++kk) {
      int c = 4 * kk + coff;
      v2f bb;  bb.x  = xb[(size_t)(c + 0) * N_ + mcol];
               bb.y  = xb[(size_t)(c + 1) * N_ + mcol];
      v2f aa1; aa1.x = a1[2 * kk]; aa1.y = a1[2 * kk + 1];
      v2f aa2; aa2.x = a2[2 * kk]; aa2.y = a2[2 * kk + 1];
      acc1 = WMMA_F32(aa1, bb, acc1);
      acc2 = WMMA_F32(aa2, bb, acc2);
    }
    const float x2v = x2[(size_t)b * N_ + mcol];
#pragma unroll
    for (int v = 0; v < 8; ++v) {
      smw[(rbase + v) * 17 + lm]      = 2.f * acc1[v] - x2v;
      smw[(16 + rbase + v) * 17 + lm] = 2.f * acc2[v] - x2v;
    }
    WAVE_LDS_FENCE();   // writes (all lanes of this wave) -> visible to reads
    // Each lane scans one of the wave's 32 rows; sorted-descending top-9.
#pragma unroll
    for (int j = 0; j < 16; ++j) {
      float v = smw[lane * 17 + j];
      if (v > tv[K_ - 1]) {
        int idx = m0 + j;
        int p = K_ - 1;
#pragma unroll
        for (int q = K_ - 2; q >= 0; --q) {
          if (v > tv[q]) { tv[q + 1] = tv[q]; ti[q + 1] = ti[q]; p = q; }
        }
        tv[p] = v; ti[p] = idx;
      }
    }
    WAVE_LDS_FENCE();   // reads complete before next tile overwrites (WAR)
  }

  int* np = nn + ((size_t)b * N_ + n0 + lane) * K_;
#pragma unroll
  for (int q = 0; q < K_; ++q) np[q] = ti[q];
}

// ---------------------------------------------------------------------------
// Kernel 3: h[b, 0:64, n] = x ; h[b, 64:128, n] = max_k (x[.,idx_k] - x[.,n])
// ---------------------------------------------------------------------------
__global__ __launch_bounds__(256) void k_gather(const float* __restrict__ x,
                                                const int* __restrict__ nn,
                                                float* __restrict__ h) {
  const int lane = threadIdx.x & 31;   // n within 32-wide tile
  const int cg   = threadIdx.x >> 5;   // channel group 0..7
  const int b = blockIdx.y;
  const int n = blockIdx.x * 32 + lane;
  const int* ip = nn + ((size_t)b * N_ + n) * K_;
  int idx[K_];
#pragma unroll
  for (int q = 0; q < K_; ++q) idx[q] = ip[q];
  const float* xb = x + (size_t)b * C_ * N_;
  float* hb = h + (size_t)b * (2 * C_) * N_;
  for (int c = cg; c < C_; c += 8) {
    float xi = xb[(size_t)c * N_ + n];
    float m = -1e30f;
#pragma unroll
    for (int q = 0; q < K_; ++q) {
      float xj = xb[(size_t)c * N_ + idx[q]];
      m = fmaxf(m, xj - xi);
    }
    hb[(size_t)c * N_ + n] = xi;
    hb[(size_t)(C_ + c) * N_ + n] = m;
  }
}

// ---------------------------------------------------------------------------
// Kernel 4: out[b,o,n] = relu(W[o,:] . h[b,:,n] + bias[o]) via fp32 WMMA.
// One wave: 16 output rows x 64 columns (4 accumulators share the W A-tile).
// ---------------------------------------------------------------------------
__global__ __launch_bounds__(256) void k_gemm(const float* __restrict__ Wm,
                                              const float* __restrict__ bias,
                                              const float* __restrict__ h,
                                              float* __restrict__ out) {
  const int lane = threadIdx.x & 31;
  const int wave = threadIdx.x >> 5;
  const int w    = blockIdx.x * 8 + wave;     // 0..4095
  const int b    = w >> 9;                    // / 512
  const int rem  = w & 511;
  const int o0   = (rem >> 6) * 16;           // 8 output tiles
  const int n0   = (rem & 63) * 64;           // 64 column chunks
  const int lm   = lane & 15;
  const int coff = (lane < 16) ? 0 : 2;
  const float* hb = h + (size_t)b * (2 * C_) * N_;
  const int orow = o0 + lm;

  v8f acc0 = {}, acc1 = {}, acc2 = {}, acc3 = {};
#pragma unroll 4
  for (int kk = 0; kk < 32; ++kk) {
    int c = 4 * kk + coff;
    v2f aa; aa.x = Wm[(size_t)orow * (2 * C_) + c];
            aa.y = Wm[(size_t)orow * (2 * C_) + c + 1];
    v2f b0; b0.x = hb[(size_t)(c + 0) * N_ + n0 + lm];
            b0.y = hb[(size_t)(c + 1) * N_ + n0 + lm];
    v2f b1; b1.x = hb[(size_t)(c + 0) * N_ + n0 + 16 + lm];
            b1.y = hb[(size_t)(c + 1) * N_ + n0 + 16 + lm];
    v2f b2; b2.x = hb[(size_t)(c + 0) * N_ + n0 + 32 + lm];
            b2.y = hb[(size_t)(c + 1) * N_ + n0 + 32 + lm];
    v2f b3; b3.x = hb[(size_t)(c + 0) * N_ + n0 + 48 + lm];
            b3.y = hb[(size_t)(c + 1) * N_ + n0 + 48 + lm];
    acc0 = WMMA_F32(aa, b0, acc0);
    acc1 = WMMA_F32(aa, b1, acc1);
    acc2 = WMMA_F32(aa, b2, acc2);
    acc3 = WMMA_F32(aa, b3, acc3);
  }

  const int rbase = (lane < 16) ? 0 : 8;
  float* ob = out + (size_t)b * O_ * N_;
#pragma unroll
  for (int v = 0; v < 8; ++v) {
    float bv = bias[o0 + rbase + v];
    float r0 = fmaxf(acc0[v] + bv, 0.f);
    float r1 = fmaxf(acc1[v] + bv, 0.f);
    float r2 = fmaxf(acc2[v] + bv, 0.f);
    float r3 = fmaxf(acc3[v] + bv, 0.f);
    size_t row = (size_t)(o0 + rbase + v) * N_;
    ob[row + n0 + lm]      = r0;
    ob[row + n0 + 16 + lm] = r1;
    ob[row + n0 + 32 + lm] = r2;
    ob[row + n0 + 48 + lm] = r3;
  }
}

// ---------------------------------------------------------------------------
extern "C" void kernel_launch(void* const* d_in, const int* in_sizes, int n_in,
                              void* d_out, int out_size, void* d_ws, size_t ws_size,
                              hipStream_t stream) {
  (void)in_sizes; (void)n_in; (void)out_size; (void)ws_size;
  const float* x  = (const float*)d_in[0];   // [B, C, N]
  const float* Wm = (const float*)d_in[1];   // [O, 2C]
  const float* bv = (const float*)d_in[2];   // [O]
  float* out = (float*)d_out;                // [B, O, N]

  char* ws = (char*)d_ws;
  float* x2 = (float*)ws;                                            // B*N f32
  int*   nn = (int*)(ws + (size_t)B_ * N_ * 4);                      // B*N*9 i32
  float* h  = (float*)(ws + (size_t)B_ * N_ * 4 +
                            (size_t)B_ * N_ * K_ * 4);               // B*2C*N f32

  k_x2    <<<dim3((B_ * N_) / 256), dim3(256), 0, stream>>>(x, x2);
  k_knn   <<<dim3(N_ / 256, B_),    dim3(256), 0, stream>>>(x, x2, nn);
  k_gather<<<dim3(N_ / 32, B_),     dim3(256), 0, stream>>>(x, nn, h);
  k_gemm  <<<dim3(512),             dim3(256), 0, stream>>>(Wm, bv, h, out);
}